// GDNAttentionLayer_31911607009297
// MI455X (gfx1250) — compile-verified
//
#include <hip/hip_runtime.h>
#include <math.h>

// GDN attention layer for MI455X (gfx1250, wave32).
// N=50000 nodes, exactly 32 contiguous incoming edges per node (dst=repeat(arange(N),32)),
// DIM=128, 4 heads x 32.
//
// Roofline: the V-row gather (E*512B = 820MB) is L2-resident (V = 25.6MB << 192MB L2)
// and dominates; GEMMs (6.5 GFLOP) are made cheap via full-precision
// V_WMMA_F32_16X16X4_F32. sq/sk are computed via a folded 128x16 effective weight
// (exact algebra: sq[n,h] = emb[n] . (W_q[:,h*32:(h+1)*32] @ a_q[h])), fused into the
// V GEMM so emb is read from HBM exactly once.

#define N_NODES 50000
#define KEDGE   32
#define DIM     128
#define HEADS   4
#define DHEAD   32

typedef __attribute__((ext_vector_type(2))) float v2f;
typedef __attribute__((ext_vector_type(8))) float v8f;

// ---------------------------------------------------------------------------
// Stage 0: fold attn_vec into W_q / W_k.
// weff[i][h]   = sum_d W_q[i][h*32+d] * a_q[h][d]   (cols 0..3)
// weff[i][4+h] = sum_d W_k[i][h*32+d] * a_k[h][d]   (cols 4..7)
// cols 8..15 zero-padded so the skinny GEMM is a clean 128x16 WMMA tile.
// ---------------------------------------------------------------------------
__global__ void build_weff_kernel(const float* __restrict__ Wq,
                                  const float* __restrict__ Wk,
                                  const float* __restrict__ av,
                                  float* __restrict__ weff) {
    int i = threadIdx.x;  // 0..127 : row of W (EMB dim)
    for (int h = 0; h < HEADS; ++h) {
        const float* aq = av + h * 2 * DHEAD;
        const float* ak = aq + DHEAD;
        float sq = 0.f, sk = 0.f;
        for (int d = 0; d < DHEAD; ++d) {
            sq = fmaf(Wq[i * DIM + h * DHEAD + d], aq[d], sq);
            sk = fmaf(Wk[i * DIM + h * DHEAD + d], ak[d], sk);
        }
        weff[i * 16 + h]     = sq;
        weff[i * 16 + 4 + h] = sk;
    }
    for (int c = 8; c < 16; ++c) weff[i * 16 + c] = 0.f;
}

// ---------------------------------------------------------------------------
// Stage 1 (fused): per 16-node block:
//   waves 0..7 : V(16x128) = [x||emb](16x256) @ W_v(256x128)   (64 WMMAs deep)
//   wave  8    : sqsk(16x8) = emb(16x128) @ weff(128x16)        (32 WMMAs deep)
// A tile staged once in LDS (row stride 258 -> conflict-free float2 reads).
// WMMA f32 16x16x4 layouts per ISA: A lanes 0-15 hold K=k0,k0+1, lanes 16-31
// hold K=k0+2,k0+3; B mirrored with N striped across lanes; C/D VGPR v -> M=v
// (lanes 0-15) / M=v+8 (lanes 16-31).
// ---------------------------------------------------------------------------
__global__ void __launch_bounds__(288) vgemm_sqsk_kernel(const float* __restrict__ x,
                                                         const float* __restrict__ emb,
                                                         const float* __restrict__ Wv,
                                                         const float* __restrict__ weff,
                                                         float* __restrict__ Vout,
                                                         float* __restrict__ sqsk) {
    __shared__ float Atile[16 * 258];
    int bx  = blockIdx.x;
    int tid = threadIdx.x;

    // Cooperative coalesced load of the 16x256 A tile (x || emb).
    for (int i = tid; i < 16 * 256; i += 288) {
        int r = i >> 8, c = i & 255;
        float v = (c < DIM) ? x[(size_t)(bx * 16 + r) * DIM + c]
                            : emb[(size_t)(bx * 16 + r) * DIM + (c - DIM)];
        Atile[r * 258 + c] = v;
    }
    __syncthreads();

    int wave = tid >> 5;
    int lane = tid & 31;
    int half = lane >> 4;
    int l15  = lane & 15;

    if (wave < 8) {
        // ---- V = [x||emb] @ W_v, one 16x16 output tile per wave ----
        int colBase = wave * 16;
        v8f acc = {};
        for (int k0 = 0; k0 < 256; k0 += 4) {
            int kb = k0 + half * 2;
            v2f a, b;
            a.x = Atile[l15 * 258 + kb];
            a.y = Atile[l15 * 258 + kb + 1];
            b.x = Wv[(size_t)kb * DIM + colBase + l15];
            b.y = Wv[(size_t)(kb + 1) * DIM + colBase + l15];
            acc = __builtin_amdgcn_wmma_f32_16x16x4_f32(
                false, a, false, b, (short)0, acc, false, false);
        }
        for (int v = 0; v < 8; ++v) {
            int row = bx * 16 + v + half * 8;
            Vout[(size_t)row * DIM + colBase + l15] = acc[v];
        }
    } else {
        // ---- sqsk = emb @ weff (emb is cols 128..255 of the LDS A tile) ----
        v8f acc = {};
        for (int k0 = 0; k0 < DIM; k0 += 4) {
            int kb = k0 + half * 2;
            v2f a, b;
            a.x = Atile[l15 * 258 + 128 + kb];
            a.y = Atile[l15 * 258 + 128 + kb + 1];
            b.x = weff[kb * 16 + l15];
            b.y = weff[(kb + 1) * 16 + l15];
            acc = __builtin_amdgcn_wmma_f32_16x16x4_f32(
                false, a, false, b, (short)0, acc, false, false);
        }
        if (l15 < 8) {
            for (int v = 0; v < 8; ++v) {
                int row = bx * 16 + v + half * 8;
                sqsk[(size_t)row * 8 + l15] = acc[v];
            }
        }
    }
}

// ---------------------------------------------------------------------------
// Stage 2: one wave per node. Lane j computes edge j's score + softmax weight
// (leaky-relu -> max-shifted softmax across the wave). Then each lane owns 4
// output columns: 32 iterations of saddr-form global_load_b128 (512B coalesced
// per edge row, L2-resident), weighted accumulate, ELU, single-wave LayerNorm.
// ---------------------------------------------------------------------------
__global__ void __launch_bounds__(32) attn_agg_kernel(const int*   __restrict__ src,
                                                      const float* __restrict__ sqsk,
                                                      const float* __restrict__ V,
                                                      const float* __restrict__ gamma,
                                                      const float* __restrict__ beta,
                                                      float* __restrict__ out) {
    __shared__ float s_alpha[HEADS][33];  // padded: conflict-free broadcast reads

    int n    = blockIdx.x;
    int lane = threadIdx.x;

    // --- edge scores: lane = edge j ---
    int s = src[(size_t)n * KEDGE + lane];
    const float* sq_n = sqsk + (size_t)n * 8;      // cols 0..3
    const float* sk_s = sqsk + (size_t)s * 8 + 4;  // cols 4..7
    float e[HEADS];
#pragma unroll
    for (int h = 0; h < HEADS; ++h) {
        float t = sq_n[h] + sk_s[h];
        e[h] = t > 0.f ? t : 0.2f * t;  // leaky relu
    }
#pragma unroll
    for (int h = 0; h < HEADS; ++h) {
        float m = e[h];
        for (int off = 16; off >= 1; off >>= 1)
            m = fmaxf(m, __shfl_xor(m, off, 32));
        float ex = __expf(e[h] - m);
        float ssum = ex;
        for (int off = 16; off >= 1; off >>= 1)
            ssum += __shfl_xor(ssum, off, 32);
        s_alpha[h][lane] = ex / (ssum + 1e-8f);
    }
    __syncthreads();

    // --- weighted gather: lane owns columns c0..c0+3 (all in head hsel) ---
    int c0   = lane * 4;
    int hsel = lane >> 3;
    const float* aptr = &s_alpha[hsel][0];
    float a0 = 0.f, a1 = 0.f, a2 = 0.f, a3 = 0.f;
#pragma unroll
    for (int j = 0; j < KEDGE; ++j) {
        int sj = __builtin_amdgcn_readfirstlane(__shfl(s, j, 32));
        float alpha = aptr[j];  // LDS broadcast (4 distinct banks)
        const float4 v = *(const float4*)(V + (size_t)sj * DIM + c0);
        a0 = fmaf(alpha, v.x, a0);
        a1 = fmaf(alpha, v.y, a1);
        a2 = fmaf(alpha, v.z, a2);
        a3 = fmaf(alpha, v.w, a3);
    }

    // --- ELU ---
    float z0 = a0 > 0.f ? a0 : (__expf(a0) - 1.f);
    float z1 = a1 > 0.f ? a1 : (__expf(a1) - 1.f);
    float z2 = a2 > 0.f ? a2 : (__expf(a2) - 1.f);
    float z3 = a3 > 0.f ? a3 : (__expf(a3) - 1.f);

    // --- LayerNorm over 128 columns (single-wave reduction) ---
    float s1 = z0 + z1 + z2 + z3;
    float s2 = z0 * z0 + z1 * z1 + z2 * z2 + z3 * z3;
    for (int off = 16; off >= 1; off >>= 1) {
        s1 += __shfl_xor(s1, off, 32);
        s2 += __shfl_xor(s2, off, 32);
    }
    float mu  = s1 * (1.f / DIM);
    float var = s2 * (1.f / DIM) - mu * mu;
    float inv = rsqrtf(var + 1e-5f);

    const float4 g = *(const float4*)(gamma + c0);
    const float4 b = *(const float4*)(beta + c0);
    float4 o;
    o.x = (z0 - mu) * inv * g.x + b.x;
    o.y = (z1 - mu) * inv * g.y + b.y;
    o.z = (z2 - mu) * inv * g.z + b.z;
    o.w = (z3 - mu) * inv * g.w + b.w;
    *(float4*)(out + (size_t)n * DIM + c0) = o;
}

// ---------------------------------------------------------------------------
extern "C" void kernel_launch(void* const* d_in, const int* in_sizes, int n_in,
                              void* d_out, int out_size, void* d_ws, size_t ws_size,
                              hipStream_t stream) {
    const float* x     = (const float*)d_in[0];
    const float* emb   = (const float*)d_in[1];
    const int*   eidx  = (const int*)  d_in[2];
    const float* Wq    = (const float*)d_in[3];
    const float* Wk    = (const float*)d_in[4];
    const float* Wv    = (const float*)d_in[5];
    const float* av    = (const float*)d_in[6];
    const float* gamma = (const float*)d_in[7];
    const float* beta  = (const float*)d_in[8];
    float* out = (float*)d_out;

    // Workspace layout: weff(128*16 f32) | sqsk(N*8 f32) | V(N*128 f32)
    char*  ws   = (char*)d_ws;
    float* weff = (float*)ws;
    float* sqsk = (float*)(ws + 128 * 16 * sizeof(float));
    float* V    = (float*)(ws + 128 * 16 * sizeof(float)
                              + (size_t)N_NODES * 8 * sizeof(float));

    const int* src = eidx;  // edge_index[0] = src; dst is repeat(arange(N),32)

    build_weff_kernel<<<1, 128, 0, stream>>>(Wq, Wk, av, weff);
    vgemm_sqsk_kernel<<<N_NODES / 16, 288, 0, stream>>>(x, emb, Wv, weff, V, sqsk);
    attn_agg_kernel<<<N_NODES, 32, 0, stream>>>(src, sqsk, V, gamma, beta, out);
}